// DistanceLoss_40647570489607
// MI455X (gfx1250) — compile-verified
//
#include <hip/hip_runtime.h>

// ---------------------------------------------------------------------------
// DistanceLoss on MI455X (gfx1250).
// Memory-bound (33.6 MB / 23.3 TB/s ~= 1.4us floor). Data path: TDM
// tensor_load_to_lds moves 16x16 fp32 tiles straight into LDS (double
// buffered, s_wait_tensorcnt / s_wait_dscnt), stencil computed from LDS,
// hierarchical deterministic reduction.
// ---------------------------------------------------------------------------

typedef unsigned int u32x4 __attribute__((ext_vector_type(4)));
typedef int          i32x8 __attribute__((ext_vector_type(8)));
typedef int          i32x4 __attribute__((ext_vector_type(4)));

#define IMG_W 2048
#define IMG_H 2048
#define PK 16                       // patch kernel size
#define NPX 128                     // patches per axis (2048/16)
#define NPATCH (NPX * NPX)          // 16384
#define WAVES_PER_BLOCK 8
#define NBLOCKS 512
#define TOTAL_WAVES (NBLOCKS * WAVES_PER_BLOCK)        // 4096
#define PATCHES_PER_WAVE (NPATCH / TOTAL_WAVES)        // 4
#define N_TERMS (NPATCH * 14 * 14 * 8)                 // 25690112

// Issue one TDM load of a 16x16 fp32 tile (row stride 2048 elements) from
// global address `gaddr` into LDS byte offset `lds_byte`.
// D# layout per CDNA5 ISA ch. 7/8 (async_tensor doc §8).
__device__ __forceinline__ void tdm_load_tile16x16_f32(unsigned long long gaddr,
                                                       unsigned lds_byte) {
  u32x4 g0;
  g0[0] = 1u;                                      // count=1 (valid), user desc
  g0[1] = lds_byte;                                // lds_addr [63:32]
  g0[2] = (unsigned)(gaddr & 0xFFFFFFFFull);       // global_addr[31:0]
  g0[3] = (unsigned)((gaddr >> 32) & 0x1FFFFFFull) // global_addr[56:32]
          | (2u << 30);                            // type=2 ("image")

  i32x8 g1;
  g1[0] = (int)(2u << 16);           // data_size = 4 bytes
  g1[1] = (int)(2048u << 16);        // tensor_dim0 = 2048 (low16 @ bits 63:48)
  g1[2] = (int)(2048u << 16);        // dim0 hi16 = 0 ; tensor_dim1 = 2048 lo16
  g1[3] = (int)(16u << 16);          // dim1 hi16 = 0 ; tile_dim0 = 16
  g1[4] = 16;                        // tile_dim1 = 16 ; tile_dim2 = 0
  g1[5] = 2048;                      // tensor_dim0_stride[31:0] = 2048 elements
  g1[6] = 0;                         // stride hi ; tensor_dim1_stride lo16 = 0
  g1[7] = 64;                        // tensor_dim1_stride[47:16] -> 2048*2048

  i32x4 gz  = {0, 0, 0, 0};          // groups 2/3 unused for 2D tile
  i32x8 gz8 = {0, 0, 0, 0, 0, 0, 0, 0};
  __builtin_amdgcn_tensor_load_to_lds(g0, g1, gz, gz, gz8, 0);
}

__global__ __launch_bounds__(256) void dloss_main(const float* __restrict__ sr,
                                                  const float* __restrict__ hr,
                                                  float* __restrict__ partials) {
  // [wave][buffer][image][16*16]
  __shared__ float tile[WAVES_PER_BLOCK][2][2][PK * PK];
  __shared__ float wsum[WAVES_PER_BLOCK];

  const int tid  = threadIdx.x;
  const int lane = tid & 31;
  const int wid  = tid >> 5;
  const int gw   = blockIdx.x * WAVES_PER_BLOCK + wid;   // global wave id

  const unsigned long long srb = (unsigned long long)(uintptr_t)sr;
  const unsigned long long hrb = (unsigned long long)(uintptr_t)hr;

  float acc = 0.0f;

  auto issue = [&](int k, int buf) {
    const int p  = gw + k * TOTAL_WAVES;
    const int pr = p >> 7;        // p / 128
    const int pc = p & 127;       // p % 128
    const unsigned long long off =
        ((unsigned long long)(pr * PK) * IMG_W + (unsigned long long)(pc * PK)) *
        sizeof(float);
    // Flat shared-memory addresses: low 32 bits are the LDS byte offset.
    const unsigned lds_s = (unsigned)(uintptr_t)(&tile[wid][buf][0][0]);
    const unsigned lds_h = (unsigned)(uintptr_t)(&tile[wid][buf][1][0]);
    tdm_load_tile16x16_f32(srb + off, lds_s);
    tdm_load_tile16x16_f32(hrb + off, lds_h);
  };

  issue(0, 0);
  for (int k = 0; k < PATCHES_PER_WAVE; ++k) {
    const int buf = k & 1;
    if (k + 1 < PATCHES_PER_WAVE) {
      // WAR guard: ensure previous iteration's ds_loads from the buffer we
      // are about to refill have fully completed before the DMA can write it.
      asm volatile("s_wait_dscnt 0" ::: "memory");
      issue(k + 1, buf ^ 1);                  // prefetch next patch (2 DMAs)
      __builtin_amdgcn_s_wait_tensorcnt(2);   // current patch's 2 DMAs done
    } else {
      __builtin_amdgcn_s_wait_tensorcnt(0);
    }

    const float* s = &tile[wid][buf][0][0];
    const float* h = &tile[wid][buf][1][0];

    // 14x14 interior pixels, 32 lanes -> up to 7 pixels per lane.
    for (int t = lane; t < 14 * 14; t += 32) {
      const int i   = 1 + t / 14;
      const int j   = 1 + t % 14;
      const int idx = i * PK + j;
      const float cs = s[idx];
      const float ch = h[idx];
      // neighbor offsets in a 16-wide row-major patch
      const float e  = fabsf(fabsf(cs - s[idx + 1])  - fabsf(ch - h[idx + 1]));
      const float nw = fabsf(fabsf(cs - s[idx - 17]) - fabsf(ch - h[idx - 17]));
      const float ne = fabsf(fabsf(cs - s[idx - 15]) - fabsf(ch - h[idx - 15]));
      const float nn = fabsf(fabsf(cs - s[idx - 16]) - fabsf(ch - h[idx - 16]));
      const float sw = fabsf(fabsf(cs - s[idx + 15]) - fabsf(ch - h[idx + 15]));
      const float se = fabsf(fabsf(cs - s[idx + 17]) - fabsf(ch - h[idx + 17]));
      const float ss = fabsf(fabsf(cs - s[idx + 16]) - fabsf(ch - h[idx + 16]));
      acc += 2.0f * e + nw + ne + nn + sw + se + ss;   // E appears twice in ref
    }
  }

  // wave32 butterfly reduction
  for (int m = 16; m >= 1; m >>= 1)
    acc += __shfl_xor(acc, m, 32);
  if (lane == 0) wsum[wid] = acc;
  __syncthreads();
  if (tid == 0) {
    float b = 0.0f;
#pragma unroll
    for (int w = 0; w < WAVES_PER_BLOCK; ++w) b += wsum[w];
    partials[blockIdx.x] = b;
  }
}

__global__ __launch_bounds__(512) void dloss_reduce(const float* __restrict__ partials,
                                                    float* __restrict__ out) {
  __shared__ float red[NBLOCKS];
  const int t = threadIdx.x;
  red[t] = partials[t];
  __syncthreads();
  for (int s2 = NBLOCKS / 2; s2 >= 1; s2 >>= 1) {
    if (t < s2) red[t] += red[t + s2];
    __syncthreads();
  }
  if (t == 0) out[0] = red[0] * (1.0f / (float)N_TERMS);
}

extern "C" void kernel_launch(void* const* d_in, const int* in_sizes, int n_in,
                              void* d_out, int out_size, void* d_ws, size_t ws_size,
                              hipStream_t stream) {
  const float* sr = (const float*)d_in[0];
  const float* hr = (const float*)d_in[1];
  float* ws = (float*)d_ws;           // NBLOCKS floats of scratch (2 KB)

  dloss_main<<<NBLOCKS, 256, 0, stream>>>(sr, hr, ws);
  dloss_reduce<<<1, NBLOCKS, 0, stream>>>(ws, (float*)d_out);
}